// ConditionalRandomField_49941879718002
// MI455X (gfx1250) — compile-verified
//
#include <hip/hip_runtime.h>

// ---------------------------------------------------------------------------
// CRF log-likelihood for B=128, S=4096, T=64 on gfx1250 (MI455X).
//
// Denominator (forward algorithm) rewritten in the scaled-exp domain:
//   m      = max_i alpha[i]                 (per sequence)
//   V      = exp(alpha - m)                 (16 x 64 per block)
//   s      = V @ E,  E = exp(transitions)   (f32 GEMM on the WMMA pipe)
//   alpha' = m + log(s) + logit_t           (masked update)
// This turns the T^2 logsumexp into V_WMMA_F32_16X16X4_F32 work with no
// precision loss relative to the f32 reference (only exp/log rounding).
// ---------------------------------------------------------------------------

constexpr int kB = 128;
constexpr int kS = 4096;
constexpr int kT = 64;
constexpr int kVStride = 68;  // padded row stride -> conflict-free LDS access

typedef __attribute__((ext_vector_type(2))) float v2f;
typedef __attribute__((ext_vector_type(8))) float v8f;

// ---------------------------------------------------------------------------
// Kernel 1: forward scan. Block = 128 threads = 4 waves; block handles 16
// sequences (WMMA M=16); wave w owns output-tag columns [16w, 16w+16).
// alpha is carried in the WMMA C/D register layout (v8f) for all 4096 steps.
// ---------------------------------------------------------------------------
__global__ __launch_bounds__(128) void crf_forward_kernel(
    const float* __restrict__ inputs,        // [B][S][T] f32
    const unsigned char* __restrict__ mask,  // [B][S] bool (1 byte)
    const float* __restrict__ transitions,   // [T][T] f32
    const float* __restrict__ start_t,       // [T] f32
    const float* __restrict__ end_t,         // [T] f32
    float* __restrict__ den) {               // [B] f32 out
  __shared__ __align__(16) float ldsV[16 * kVStride];  // V = exp(alpha - m)
  __shared__ __align__(16) float ldsPM[16 * 4];        // per-wave partials

  const int tid = threadIdx.x;
  const int w = tid >> 5;           // wave id = output-tag tile
  const int lane = tid & 31;
  const int nlane = lane & 15;
  const int hi = lane >> 4;         // half-wave
  const int half8 = hi * 8;         // C/D row offset (rows r / r+8)
  const int kOff = hi * 2;          // A/B K offset (K pairs {0,1}/{2,3})
  const int jcol = w * 16 + nlane;  // output tag column owned by this lane
  const int b0 = blockIdx.x * 16;

  // B-fragments of E = exp(transitions): ISA f32 B layout (4x16 per frag):
  //   vgpr0: K=4kc+kOff, N=jcol ; vgpr1: K=4kc+kOff+1. Held in VGPRs forever.
  float eB0[16], eB1[16];
#pragma unroll
  for (int kc = 0; kc < 16; ++kc) {
    eB0[kc] = __expf(transitions[(kc * 4 + kOff + 0) * kT + jcol]);
    eB1[kc] = __expf(transitions[(kc * 4 + kOff + 1) * kT + jcol]);
  }
  const float endv = end_t[jcol];
  const float sttv = start_t[jcol];

  // alpha[r] = alpha[sequence b0 + r + half8][tag jcol]  (C/D layout)
  v8f alpha;
#pragma unroll
  for (int r = 0; r < 8; ++r) {
    alpha[r] = sttv + inputs[(size_t)(b0 + r + half8) * kS * kT + jcol];
  }

  for (int t = 1; t < kS; ++t) {
    // ---- per-sequence max over 64 tags: shuffle within wave, LDS across ----
    float pmax[8];
#pragma unroll
    for (int r = 0; r < 8; ++r) {
      float v = alpha[r];
      v = fmaxf(v, __shfl_xor(v, 8, 32));  // masks <= 8 stay in 16-lane group
      v = fmaxf(v, __shfl_xor(v, 4, 32));
      v = fmaxf(v, __shfl_xor(v, 2, 32));
      v = fmaxf(v, __shfl_xor(v, 1, 32));
      pmax[r] = v;
    }
    if (nlane == 0) {
#pragma unroll
      for (int r = 0; r < 8; ++r) ldsPM[(r + half8) * 4 + w] = pmax[r];
    }
    __syncthreads();

    float mrow[8];
#pragma unroll
    for (int r = 0; r < 8; ++r) {
      const float4 pm = *reinterpret_cast<const float4*>(&ldsPM[(r + half8) * 4]);
      mrow[r] = fmaxf(fmaxf(pm.x, pm.y), fmaxf(pm.z, pm.w));
    }

    // ---- V = exp(alpha - m) staged to LDS in [sequence][tag] order ----
#pragma unroll
    for (int r = 0; r < 8; ++r)
      ldsV[(r + half8) * kVStride + jcol] = __expf(alpha[r] - mrow[r]);
    __syncthreads();

    // ---- s = V(16x64) @ E(64x64), this wave's 16-column tile ----
    v8f acc = {};
#pragma unroll
    for (int kc = 0; kc < 16; ++kc) {
      // A-fragment (f32 16x4): lane row = nlane, K = 4kc+kOff {+0,+1}
      const v2f a =
          *reinterpret_cast<const v2f*>(&ldsV[nlane * kVStride + kc * 4 + kOff]);
      v2f b;
      b.x = eB0[kc];
      b.y = eB1[kc];
      acc = __builtin_amdgcn_wmma_f32_16x16x4_f32(
          false, a, false, b, (short)0, acc, false, false);
    }

    // ---- alpha' = m + log(s) + logit_t, gated by mask (per-lane select) ----
#pragma unroll
    for (int r = 0; r < 8; ++r) {
      const int m = b0 + r + half8;
      const bool mk = mask[(size_t)m * kS + t] != 0;
      const float lg = inputs[((size_t)m * kS + t) * kT + jcol];
      const float inner = mrow[r] + __logf(acc[r]) + lg;
      alpha[r] = mk ? inner : alpha[r];
    }
  }

  // ---- den[b] = logsumexp_j(alpha + end) ----
#pragma unroll
  for (int r = 0; r < 8; ++r) alpha[r] += endv;

  float pmax[8];
#pragma unroll
  for (int r = 0; r < 8; ++r) {
    float v = alpha[r];
    v = fmaxf(v, __shfl_xor(v, 8, 32));
    v = fmaxf(v, __shfl_xor(v, 4, 32));
    v = fmaxf(v, __shfl_xor(v, 2, 32));
    v = fmaxf(v, __shfl_xor(v, 1, 32));
    pmax[r] = v;
  }
  if (nlane == 0) {
#pragma unroll
    for (int r = 0; r < 8; ++r) ldsPM[(r + half8) * 4 + w] = pmax[r];
  }
  __syncthreads();
  float mrow[8];
#pragma unroll
  for (int r = 0; r < 8; ++r) {
    const float4 pm = *reinterpret_cast<const float4*>(&ldsPM[(r + half8) * 4]);
    mrow[r] = fmaxf(fmaxf(pm.x, pm.y), fmaxf(pm.z, pm.w));
  }
  __syncthreads();  // ldsPM reused for partial sums below
#pragma unroll
  for (int r = 0; r < 8; ++r) {
    float p = __expf(alpha[r] - mrow[r]);
    p += __shfl_xor(p, 8, 32);
    p += __shfl_xor(p, 4, 32);
    p += __shfl_xor(p, 2, 32);
    p += __shfl_xor(p, 1, 32);
    if (nlane == 0) ldsPM[(r + half8) * 4 + w] = p;
  }
  __syncthreads();
  if (w == 0 && nlane == 0) {
#pragma unroll
    for (int r = 0; r < 8; ++r) {
      const float4 pm = *reinterpret_cast<const float4*>(&ldsPM[(r + half8) * 4]);
      den[b0 + r + half8] = mrow[r] + __logf(pm.x + pm.y + pm.z + pm.w);
    }
  }
}

// ---------------------------------------------------------------------------
// Kernel 2: joint (numerator) score. One block per sequence.
// ---------------------------------------------------------------------------
__global__ __launch_bounds__(256) void crf_numerator_kernel(
    const float* __restrict__ inputs, const int* __restrict__ tags,
    const unsigned char* __restrict__ mask,
    const float* __restrict__ transitions, const float* __restrict__ start_t,
    const float* __restrict__ end_t, float* __restrict__ num) {
  const int b = blockIdx.x;
  const int tid = threadIdx.x;
  const int* tg = tags + (size_t)b * kS;
  const unsigned char* mk = mask + (size_t)b * kS;
  const float* lg = inputs + (size_t)b * kS * kT;

  float acc = 0.f;
  int cnt = 0;
  for (int t = tid; t < kS; t += 256) {
    const bool m = mk[t] != 0;
    cnt += m ? 1 : 0;
    if (m) {
      if (t >= 1) acc += transitions[tg[t - 1] * kT + tg[t]];        // * maskf[1:]
      if (t <= kS - 2) acc += lg[(size_t)t * kT + tg[t]];            // * maskf[:-1]
    }
  }
  __shared__ float sf[256];
  __shared__ int si[256];
  sf[tid] = acc;
  si[tid] = cnt;
  __syncthreads();
  for (int s = 128; s > 0; s >>= 1) {
    if (tid < s) {
      sf[tid] += sf[tid + s];
      si[tid] += si[tid + s];
    }
    __syncthreads();
  }
  if (tid == 0) {
    const int last_idx = si[0] - 1;  // lengths >= S/2 >= 1
    const int last_tag = tg[last_idx];
    float total = sf[0] + start_t[tg[0]] + end_t[last_tag];
    if (mk[kS - 1]) total += lg[(size_t)(kS - 1) * kT + last_tag];
    num[b] = total;
  }
}

// ---------------------------------------------------------------------------
// Kernel 3: out = sum_b (num[b] - den[b])
// ---------------------------------------------------------------------------
__global__ __launch_bounds__(128) void crf_reduce_kernel(
    const float* __restrict__ den, const float* __restrict__ num,
    float* __restrict__ out) {
  __shared__ float s[kB];
  const int i = threadIdx.x;
  s[i] = num[i] - den[i];
  __syncthreads();
  for (int st = kB / 2; st > 0; st >>= 1) {
    if (i < st) s[i] += s[i + st];
    __syncthreads();
  }
  if (i == 0) out[0] = s[0];
}

extern "C" void kernel_launch(void* const* d_in, const int* in_sizes, int n_in,
                              void* d_out, int out_size, void* d_ws,
                              size_t ws_size, hipStream_t stream) {
  (void)in_sizes; (void)n_in; (void)out_size; (void)ws_size;
  const float* inputs = (const float*)d_in[0];
  const int* tags = (const int*)d_in[1];
  const unsigned char* mask = (const unsigned char*)d_in[2];  // jnp bool = u8
  const float* transitions = (const float*)d_in[3];
  const float* start_t = (const float*)d_in[4];
  const float* end_t = (const float*)d_in[5];
  float* out = (float*)d_out;
  float* den = (float*)d_ws;       // [128]
  float* num = den + kB;           // [128]

  crf_forward_kernel<<<dim3(kB / 16), dim3(128), 0, stream>>>(
      inputs, mask, transitions, start_t, end_t, den);
  crf_numerator_kernel<<<dim3(kB), dim3(256), 0, stream>>>(
      inputs, tags, mask, transitions, start_t, end_t, num);
  crf_reduce_kernel<<<dim3(1), dim3(kB), 0, stream>>>(den, num, out);
}